// MatrixHyperlayer_60799557042178
// MI455X (gfx1250) — compile-verified
//
#include <hip/hip_runtime.h>

// ---------------------------------------------------------------------------
// MatrixHyperlayer scatter  ==  sparse (20000x20000, 1.2M nnz) @ dense (20000x128)
// Strategy: bin nonzeros by 16-row output tile, then tile-owned WMMA-f32 SpMM
// (no output atomics, exact f32 math, L2-resident traffic).
// ---------------------------------------------------------------------------

#define IN_NUM   20000
#define OUT_NUM  20000
#define KNUM     30000
#define NPTS     20            // 4 neighbors + 16 sampled
#define FEAT     128
#define NTILES   1250          // 20000 / 16
#define E_TOTAL  (KNUM * NPTS * 2)   // 1,200,000 directed nonzeros (fixed)

typedef __attribute__((ext_vector_type(2))) float v2f;
typedef __attribute__((ext_vector_type(8))) float v8f;

// ---------------------------------------------------------------------------
// Per-k math, matching the reference op-for-op (same multiply ordering so the
// floor/ceil integer indices match the JAX reference bit-for-bit).
// ---------------------------------------------------------------------------
__device__ __forceinline__ void compute_points(const float4 p,
                                               const float2* __restrict__ su16,
                                               int* ix, int* iy, float* vv)
{
    const float EPSf = 1e-6f;
    float sgx = 1.f / (1.f + expf(-p.x));
    float sgy = 1.f / (1.f + expf(-p.y));
    float mx = sgx * (float)(OUT_NUM - 1);
    float my = sgy * (float)(IN_NUM - 1);

    float sp    = p.z + 2.0f;                                  // SIGMA_BOOST
    float softp = (sp > 20.f) ? sp : log1pf(expf(sp));
    float sigma = (softp + EPSf) * 20000.f * 0.2f;             // * size * SIGMA_SCALE
    float inv_std = sqrtf(1.f / (EPSf + sigma));               // same for both dims
    float val = tanhf(p.w);

    float px[NPTS], py[NPTS];
    float fx = floorf(mx), cx = ceilf(mx);
    float fy = floorf(my), cy = ceilf(my);
    // FLOOR_MASK = [[T,T],[T,F],[F,T],[F,F]]  (True -> floor)
    px[0] = fx; py[0] = fy;
    px[1] = fx; py[1] = cy;
    px[2] = cx; py[2] = fy;
    px[3] = cx; py[3] = cy;
#pragma unroll
    for (int t = 0; t < 16; ++t) {
        float2 u = su16[t];
        px[4 + t] = floorf(u.x * (1.f - EPSf) * 20000.f);
        py[4 + t] = floorf(u.y * (1.f - EPSf) * 20000.f);
    }

    float num[NPTS];
    float sum = (float)NPTS * EPSf;                            // + D*EPS
#pragma unroll
    for (int t = 0; t < NPTS; ++t) {
        float dx = (px[t] - mx) * inv_std;
        float dy = (py[t] - my) * inv_std;
        float e  = expf(-0.5f * (dx * dx + dy * dy));
        num[t] = e;
        sum += e;
    }
    float scale = val / sum;
#pragma unroll
    for (int t = 0; t < NPTS; ++t) {
        int a = (int)px[t];
        int b = (int)py[t];
        ix[t] = a;
        iy[t] = b;
        vv[t] = (a == b) ? 0.f : num[t] * scale;
    }
}

// ---------------------------------------------------------------------------
// Utility kernels
// ---------------------------------------------------------------------------
__global__ void k_zero_i32(int* __restrict__ p, int n)
{
    int i = blockIdx.x * 256 + threadIdx.x;
    if (i < n) p[i] = 0;
}

__global__ void k_zero_f32(float* __restrict__ p, int n)
{
    int i = blockIdx.x * 256 + threadIdx.x;
    if (i < n) p[i] = 0.f;
}

// Count nonzeros per 16-row output tile (LDS histogram -> global flush).
__global__ void __launch_bounds__(256) k_count(const float* __restrict__ params,
                                               const float* __restrict__ su,
                                               int* __restrict__ counts)
{
    __shared__ int hist[NTILES];
    for (int t = threadIdx.x; t < NTILES; t += 256) hist[t] = 0;
    __syncthreads();

    int k = blockIdx.x * 256 + threadIdx.x;
    if (k < KNUM) {
        int ix[NPTS], iy[NPTS];
        float vv[NPTS];
        float4 p = reinterpret_cast<const float4*>(params)[k];
        const float2* su16 = reinterpret_cast<const float2*>(su) + (size_t)k * 16;
        compute_points(p, su16, ix, iy, vv);
#pragma unroll
        for (int t = 0; t < NPTS; ++t) {
            atomicAdd(&hist[ix[t] >> 4], 1);   // entry (i=ix, j=iy)
            atomicAdd(&hist[iy[t] >> 4], 1);   // symmetric (i=iy, j=ix)
        }
    }
    __syncthreads();
    for (int t = threadIdx.x; t < NTILES; t += 256)
        if (hist[t]) atomicAdd(&counts[t], hist[t]);
}

// Serial exclusive scan over 1250 tiles (trivial cost).
__global__ void k_scan(const int* __restrict__ counts,
                       int* __restrict__ offsets,
                       int* __restrict__ cursor)
{
    int run = 0;
    for (int t = 0; t < NTILES; ++t) {
        offsets[t] = run;
        cursor[t]  = run;
        run += counts[t];
    }
}

// Recompute entries and scatter them into their bins.
// bjr packs (local_row<<16 | j); j < 20000 fits in 16 bits.
__global__ void __launch_bounds__(256) k_place(const float* __restrict__ params,
                                               const float* __restrict__ su,
                                               int* __restrict__ cursor,
                                               int* __restrict__ bjr,
                                               float* __restrict__ bv)
{
    int k = blockIdx.x * 256 + threadIdx.x;
    if (k >= KNUM) return;
    int ix[NPTS], iy[NPTS];
    float vv[NPTS];
    float4 p = reinterpret_cast<const float4*>(params)[k];
    const float2* su16 = reinterpret_cast<const float2*>(su) + (size_t)k * 16;
    compute_points(p, su16, ix, iy, vv);
#pragma unroll
    for (int t = 0; t < NPTS; ++t) {
        int i = ix[t], j = iy[t];
        float v = vv[t];
        int pos = atomicAdd(&cursor[i >> 4], 1);
        bjr[pos] = ((i & 15) << 16) | j;
        bv[pos]  = v;
        pos = atomicAdd(&cursor[j >> 4], 1);
        bjr[pos] = ((j & 15) << 16) | i;
        bv[pos]  = v;
    }
}

// ---------------------------------------------------------------------------
// WMMA SpMM: one block (8 wave32s) owns one 16-row output tile.
//   - 32 nonzeros staged per iteration: their input rows (f32) into LDS.
//   - wave w owns feature slice [16w, 16w+16); accumulates 16x16 f32 tile
//     via V_WMMA_F32_16X16X4_F32: A(16x4) one-hot v-matrix, B(4x16) row slice.
//   - exact f32 math; no output atomics (plain stores).
// LDS row pitch 136 f32: K-halves land on disjoint bank groups for B reads.
// ---------------------------------------------------------------------------
__global__ void __launch_bounds__(256) k_spmm(const float* __restrict__ input,
                                              const int* __restrict__ bjr,
                                              const float* __restrict__ bv,
                                              const int* __restrict__ offsets,
                                              const int* __restrict__ counts,
                                              float* __restrict__ out)
{
    const int tile  = blockIdx.x;
    const int begin = offsets[tile];
    const int n     = counts[tile];
    const int tid   = threadIdx.x;
    const int wave  = tid >> 5;
    const int lane  = tid & 31;
    const int nn    = lane & 15;   // A: M row / B,C: N column for this lane
    const int kg    = lane >> 4;   // half-wave select (K pair 0,1 vs 2,3; M 0-7 vs 8-15)
    const int fbase = wave * 16;

    __shared__ float rows[32][136];
    __shared__ int   srow[32];
    __shared__ int   sj[32];
    __shared__ float sval[32];

    v8f acc = {};   // 16x16 f32 accumulator tile (8 VGPRs, wave32 layout)

    for (int base = 0; base < n; base += 32) {
        __syncthreads();
        if (tid < 32) {
            int idx = base + tid;
            int jr = 255 << 16;     // local row 255: matches no lane -> A contributes 0
            float v = 0.f;
            int j = 0;              // stage row 0 harmlessly for padding
            if (idx < n) {
                jr = bjr[begin + idx];
                v  = bv[begin + idx];
                j  = jr & 0xFFFF;
            }
            srow[tid] = jr >> 16;
            sj[tid]   = j;
            sval[tid] = v;
            // gfx1250 prefetch of next chunk's metadata (stays inside ws buffer)
            __builtin_prefetch(bjr + begin + base + 32 + tid, 0, 1);
        }
        __syncthreads();
        {
            // Stage 32 gathered input rows (f32) into LDS: thread t -> row t/8, 16 feats.
            int r  = tid >> 3;
            int fg = tid & 7;
            const float4* src =
                reinterpret_cast<const float4*>(input + (size_t)sj[r] * FEAT + fg * 16);
            float4 a0 = src[0], a1 = src[1], a2 = src[2], a3 = src[3];
            float4* dst = reinterpret_cast<float4*>(&rows[r][fg * 16]);
            dst[0] = a0; dst[1] = a1; dst[2] = a2; dst[3] = a3;
        }
        __syncthreads();

#pragma unroll
        for (int kk = 0; kk < 8; ++kk) {       // 8 x (K=4) = 32 nonzeros
            int c0 = kk * 4 + 2 * kg;          // this half-wave's K pair
            v2f A, B;
            // A (16x4 f32): lanes 0-15 hold K=0,1; lanes 16-31 hold K=2,3
            A[0] = (srow[c0]     == nn) ? sval[c0]     : 0.f;
            A[1] = (srow[c0 + 1] == nn) ? sval[c0 + 1] : 0.f;
            // B (4x16 f32): VGPR0 = B[{0|2}, nn], VGPR1 = B[{1|3}, nn]
            B[0] = rows[c0][fbase + nn];
            B[1] = rows[c0 + 1][fbase + nn];
            acc = __builtin_amdgcn_wmma_f32_16x16x4_f32(
                /*neg_a=*/false, A, /*neg_b=*/false, B,
                /*c_mod=*/(short)0, acc, /*reuse_a=*/false, /*reuse_b=*/false);
        }
    }

    // Store 16x16 tile: VGPR r -> M=r (lanes 0-15) / M=r+8 (lanes 16-31), N=nn.
    const int mbase = tile * 16 + kg * 8;
#pragma unroll
    for (int r = 0; r < 8; ++r)
        out[(size_t)(mbase + r) * FEAT + fbase + nn] = acc[r];
}

// ---------------------------------------------------------------------------
// Fallback (workspace too small): zero output + f32 atomic scatter, one
// wave32 per k, lane handles 4 features.
// ---------------------------------------------------------------------------
__global__ void __launch_bounds__(256) k_scatter_atomic(const float* __restrict__ input,
                                                        const float* __restrict__ params,
                                                        const float* __restrict__ su,
                                                        float* __restrict__ out)
{
    int k    = (blockIdx.x * 256 + threadIdx.x) >> 5;   // one wave per k
    int lane = threadIdx.x & 31;
    if (k >= KNUM) return;

    int ix[NPTS], iy[NPTS];
    float vv[NPTS];
    float4 p = reinterpret_cast<const float4*>(params)[k];
    const float2* su16 = reinterpret_cast<const float2*>(su) + (size_t)k * 16;
    compute_points(p, su16, ix, iy, vv);

    for (int t = 0; t < NPTS; ++t) {
        float v = vv[t];
        if (v == 0.f) continue;                 // uniform across the wave
        int i = ix[t], j = iy[t];
        float4 rj = reinterpret_cast<const float4*>(input + (size_t)j * FEAT)[lane];
        float4 ri = reinterpret_cast<const float4*>(input + (size_t)i * FEAT)[lane];
        float* oi = out + (size_t)i * FEAT + lane * 4;
        float* oj = out + (size_t)j * FEAT + lane * 4;
        unsafeAtomicAdd(oi + 0, v * rj.x); unsafeAtomicAdd(oi + 1, v * rj.y);
        unsafeAtomicAdd(oi + 2, v * rj.z); unsafeAtomicAdd(oi + 3, v * rj.w);
        unsafeAtomicAdd(oj + 0, v * ri.x); unsafeAtomicAdd(oj + 1, v * ri.y);
        unsafeAtomicAdd(oj + 2, v * ri.z); unsafeAtomicAdd(oj + 3, v * ri.w);
    }
}

// ---------------------------------------------------------------------------
extern "C" void kernel_launch(void* const* d_in, const int* in_sizes, int n_in,
                              void* d_out, int out_size, void* d_ws, size_t ws_size,
                              hipStream_t stream)
{
    const float* input  = (const float*)d_in[0];   // (20000, 128)
    const float* params = (const float*)d_in[1];   // (30000, 4)
    const float* su     = (const float*)d_in[2];   // (30000, 16, 2)
    float* out = (float*)d_out;                    // (20000, 128)

    const size_t off_counts  = 0;
    const size_t off_offsets = 8192;
    const size_t off_cursor  = 16384;
    const size_t off_bjr     = 24576;
    const size_t off_bv      = off_bjr + (size_t)E_TOTAL * sizeof(int);
    const size_t need        = off_bv  + (size_t)E_TOTAL * sizeof(float);  // ~9.6 MB

    if (ws_size >= need) {
        int*   counts  = (int*)((char*)d_ws + off_counts);
        int*   offsets = (int*)((char*)d_ws + off_offsets);
        int*   cursor  = (int*)((char*)d_ws + off_cursor);
        int*   bjr     = (int*)((char*)d_ws + off_bjr);
        float* bv      = (float*)((char*)d_ws + off_bv);

        hipLaunchKernelGGL(k_zero_i32, dim3((NTILES + 255) / 256), dim3(256), 0, stream,
                           counts, NTILES);
        hipLaunchKernelGGL(k_count, dim3((KNUM + 255) / 256), dim3(256), 0, stream,
                           params, su, counts);
        hipLaunchKernelGGL(k_scan, dim3(1), dim3(1), 0, stream, counts, offsets, cursor);
        hipLaunchKernelGGL(k_place, dim3((KNUM + 255) / 256), dim3(256), 0, stream,
                           params, su, cursor, bjr, bv);
        hipLaunchKernelGGL(k_spmm, dim3(NTILES), dim3(256), 0, stream,
                           input, bjr, bv, offsets, counts, out);
    } else {
        int nOut = OUT_NUM * FEAT;
        hipLaunchKernelGGL(k_zero_f32, dim3((nOut + 255) / 256), dim3(256), 0, stream,
                           out, nOut);
        hipLaunchKernelGGL(k_scatter_atomic, dim3((KNUM * 32 + 255) / 256), dim3(256), 0,
                           stream, input, params, su, out);
    }
}